// Specialspmm_17171279249898
// MI455X (gfx1250) — compile-verified
//
#include <hip/hip_runtime.h>

// SpMM out[i,:] = sum_{e:(i,j)} values[e] * b[j,:]   (E=1.6M, N=100K, D=64, fp32)
//
// MI455X strategy recap:
//  * b (25.6MB) and out (25.6MB) are L2-resident (192MB L2); bottleneck is L2
//    atomic/gather throughput, not HBM (total HBM traffic ~70MB ~= 3us).
//  * Build CSR on device (3.2M int atomics + 100K scan) instead of 102.4M
//    fp32 atomics; SpMM becomes a pure gather, output written exactly once.
//  * NEW: each block's (col,val) edge slice is contiguous in CSR order ->
//    stage it into LDS with gfx1250 async global->LDS copies
//    (global_load_async_to_lds_b128 + s_wait_asynccnt), then the inner loop
//    reads edge scalars as conflict-free LDS broadcasts instead of ~100M
//    redundant global broadcast loads.

#define D_FEAT 64
#define SCAN_BLOCK 256
#define LANES_PER_ROW 16     // 16 lanes * float4 = one 64-float feature row
#define ROWS_PER_BLOCK 16    // 256 threads / 16 lanes
#define CHUNK_CAP 2048       // edges per LDS window (2 * 8KB buffers)

// ---------------------------------------------------------------- utilities
__device__ __forceinline__ int atom_inc_i32(int* p) {
  return __hip_atomic_fetch_add(p, 1, __ATOMIC_RELAXED, __HIP_MEMORY_SCOPE_AGENT);
}

__device__ __forceinline__ void atom_add_f32(float* __restrict__ p, float v) {
  __hip_atomic_fetch_add(p, v, __ATOMIC_RELAXED, __HIP_MEMORY_SCOPE_AGENT);
}

// Low 32 bits of a generic pointer derived from __shared__ = LDS byte offset
__device__ __forceinline__ unsigned lds_off_of(const void* p) {
  return (unsigned)(size_t)p;
}

// gfx1250 async copy: 16 bytes global -> LDS, tracked by ASYNCcnt.
__device__ __forceinline__ void async_copy_b128(unsigned lds_off, const void* gaddr) {
  asm volatile("global_load_async_to_lds_b128 %0, %1, off"
               :: "v"(lds_off), "v"(gaddr) : "memory");
}

__device__ __forceinline__ void wait_asynccnt0() {
  asm volatile("s_wait_asynccnt 0x0" ::: "memory");
}

// -------------------------------------------------------- 1) row histogram
__global__ __launch_bounds__(256) void hist_kernel(const int* __restrict__ rows,
                                                   int* __restrict__ counts, int E) {
  int e = blockIdx.x * blockDim.x + threadIdx.x;
  if (e < E) atom_inc_i32(&counts[rows[e]]);
}

// ------------------------------------------- 2a) per-block exclusive scan
__global__ __launch_bounds__(SCAN_BLOCK) void scan_block_kernel(
    const int* __restrict__ counts, int* __restrict__ local_scan,
    int* __restrict__ block_sums, int N) {
  __shared__ int sm[SCAN_BLOCK];
  const int i = blockIdx.x * SCAN_BLOCK + threadIdx.x;
  const int x = (i < N) ? counts[i] : 0;
  sm[threadIdx.x] = x;
  __syncthreads();
  for (int off = 1; off < SCAN_BLOCK; off <<= 1) {
    int t = (threadIdx.x >= off) ? sm[threadIdx.x - off] : 0;
    __syncthreads();
    sm[threadIdx.x] += t;
    __syncthreads();
  }
  if (i < N) local_scan[i] = sm[threadIdx.x] - x;          // exclusive
  if (threadIdx.x == SCAN_BLOCK - 1) block_sums[blockIdx.x] = sm[threadIdx.x];
}

// --------------------------------- 2b) scan of block sums (nb <= 1024, 1 WG)
__global__ __launch_bounds__(1024) void scan_sums_kernel(int* __restrict__ block_sums,
                                                         int nb) {
  __shared__ int sm[1024];
  const int x = (threadIdx.x < nb) ? block_sums[threadIdx.x] : 0;
  sm[threadIdx.x] = x;
  __syncthreads();
  for (int off = 1; off < 1024; off <<= 1) {
    int t = (threadIdx.x >= off) ? sm[threadIdx.x - off] : 0;
    __syncthreads();
    sm[threadIdx.x] += t;
    __syncthreads();
  }
  if (threadIdx.x < nb) block_sums[threadIdx.x] = sm[threadIdx.x] - x;  // exclusive
}

// ---------------------- 2c) finalize: offsets[] (CSR row ptr) + cursor copy
__global__ __launch_bounds__(256) void finalize_offsets_kernel(
    const int* __restrict__ local_scan, const int* __restrict__ block_sums,
    int* __restrict__ offsets, int* __restrict__ cursor, int N, int E) {
  const int i = blockIdx.x * blockDim.x + threadIdx.x;
  if (i < N) {
    const int o = local_scan[i] + block_sums[i / SCAN_BLOCK];
    offsets[i] = o;
    cursor[i]  = o;
  }
  if (i == 0) offsets[N] = E;
}

// ------ 3) scatter: pre-gather (col,val) into row-sorted permc/permv arrays
__global__ __launch_bounds__(256) void scatter_kernel(
    const int* __restrict__ rows, const int* __restrict__ cols,
    const float* __restrict__ vals, int* __restrict__ cursor,
    int* __restrict__ permc, float* __restrict__ permv, int E) {
  const int e = blockIdx.x * blockDim.x + threadIdx.x;
  if (e >= E) return;
  const int p = atom_inc_i32(&cursor[rows[e]]);
  permc[p] = cols[e];     // sequential reads in SpMM phase
  permv[p] = vals[e];
}

// ---- 4) gather SpMM with async-LDS staging of the contiguous edge slice
__global__ __launch_bounds__(256) void spmm_csr_lds_kernel(
    const int* __restrict__ offsets, const int* __restrict__ permc,
    const float* __restrict__ permv, const float* __restrict__ b,
    float* __restrict__ out, int N) {
  __shared__ int   sm_c[CHUNK_CAP + 4];
  __shared__ float sm_v[CHUNK_CAP + 4];

  const int row0 = blockIdx.x * ROWS_PER_BLOCK;
  const int rloc = threadIdx.x / LANES_PER_ROW;
  const int lane = threadIdx.x % LANES_PER_ROW;
  const int row  = row0 + rloc;

  const int row_hi  = (row0 + ROWS_PER_BLOCK < N) ? row0 + ROWS_PER_BLOCK : N;
  const int blk_beg = offsets[row0];     // block's edges are contiguous in CSR
  const int blk_end = offsets[row_hi];

  int beg = 0, end = 0;
  if (row < N) { beg = offsets[row]; end = offsets[row + 1]; }

  float4 acc = make_float4(0.f, 0.f, 0.f, 0.f);

  // windows aligned down to 4 edges so async b128 copies are 16B-aligned
  for (int win = blk_beg & ~3; win < blk_end; win += CHUNK_CAP) {
    const int win_hi = (win + CHUNK_CAP < blk_end) ? win + CHUNK_CAP : blk_end;
    const int n4 = ((win_hi - win) + 3) & ~3;   // permc/permv are padded in ws

    // cooperative async global->LDS burst (16B per instruction per lane)
    for (int t = threadIdx.x * 4; t < n4; t += 256 * 4) {
      async_copy_b128(lds_off_of(&sm_c[t]), permc + win + t);
      async_copy_b128(lds_off_of(&sm_v[t]), permv + win + t);
    }
    wait_asynccnt0();      // each wave drains its own ASYNCcnt ...
    __syncthreads();       // ... then block-wide visibility

    int e  = (beg > win)    ? beg : win;
    int eh = (end < win_hi) ? end : win_hi;
    for (; e < eh; ++e) {
      const int   c = sm_c[e - win];   // LDS broadcast (conflict-free)
      const float v = sm_v[e - win];
      if (e + 1 < eh) {                // prefetch next edge's b row (L2->L0)
        __builtin_prefetch(b + (size_t)sm_c[e + 1 - win] * D_FEAT + lane * 4, 0, 3);
      }
      const float4 x = *((const float4*)(b + (size_t)c * D_FEAT) + lane);
      acc.x += v * x.x;
      acc.y += v * x.y;
      acc.z += v * x.z;
      acc.w += v * x.w;
    }
    __syncthreads();       // LDS reuse fence before next window
  }

  if (row < N) *((float4*)(out + (size_t)row * D_FEAT) + lane) = acc;
}

// --------------------------- fallback: one-pass COO atomic (known-good path)
__global__ __launch_bounds__(256) void spmm_coo_atomic_kernel(
    const int* __restrict__ rows, const int* __restrict__ cols,
    const float* __restrict__ vals, const float* __restrict__ b,
    float* __restrict__ out, int E) {
  const long long tid = (long long)blockIdx.x * blockDim.x + threadIdx.x;
  const int edge = (int)(tid / LANES_PER_ROW);
  const int lane = (int)(tid % LANES_PER_ROW);
  if (edge >= E) return;
  const int   r = rows[edge];
  const int   c = cols[edge];
  const float v = vals[edge];
  const float4 x = *((const float4*)(b + (size_t)c * D_FEAT) + lane);
  float* dst = out + (size_t)r * D_FEAT + (size_t)lane * 4;
  atom_add_f32(dst + 0, v * x.x);
  atom_add_f32(dst + 1, v * x.y);
  atom_add_f32(dst + 2, v * x.z);
  atom_add_f32(dst + 3, v * x.w);
}

// --------------------------------------------------------------------- host
static inline size_t align256(size_t x) { return (x + 255) & ~(size_t)255; }

extern "C" void kernel_launch(void* const* d_in, const int* in_sizes, int n_in,
                              void* d_out, int out_size, void* d_ws, size_t ws_size,
                              hipStream_t stream) {
  // d_in[0]=indices int32[2,E]; d_in[1]=values f32[E]; d_in[2]=N (unused);
  // d_in[3]=b f32[N,64]
  const int*   indices = (const int*)d_in[0];
  const float* vals    = (const float*)d_in[1];
  const float* b       = (const float*)d_in[3];
  float*       out     = (float*)d_out;

  const int E = in_sizes[1];
  const int N = out_size / D_FEAT;
  const int* rows = indices;
  const int* cols = indices + E;

  const int nb_scan = (N + SCAN_BLOCK - 1) / SCAN_BLOCK;   // 391 for N=100K

  // Workspace layout (256B-aligned slices; permc/permv padded 16 ints so the
  // rounded-up async b128 copies never read past the slice)
  size_t off = 0;
  const size_t counts_off = off; off = align256(off + (size_t)N * 4);
  const size_t local_off  = off; off = align256(off + (size_t)N * 4);
  const size_t offs_off   = off; off = align256(off + (size_t)(N + 1) * 4);
  const size_t cursor_off = off; off = align256(off + (size_t)N * 4);
  const size_t bsums_off  = off; off = align256(off + (size_t)1024 * 4);
  const size_t permc_off  = off; off = align256(off + (size_t)(E + 16) * 4);
  const size_t permv_off  = off; off = align256(off + (size_t)(E + 16) * 4);
  const size_t ws_needed  = off;

  const int block = 256;

  if (ws_size < ws_needed || nb_scan > 1024) {
    // Fallback: one-pass COO atomic SpMM (needs zeroed output each call).
    hipMemsetAsync(d_out, 0, (size_t)out_size * sizeof(float), stream);
    const long long total = (long long)E * LANES_PER_ROW;
    const int grid = (int)((total + block - 1) / block);
    spmm_coo_atomic_kernel<<<grid, block, 0, stream>>>(rows, cols, vals, b, out, E);
    return;
  }

  char* w = (char*)d_ws;
  int*   counts = (int*)(w + counts_off);
  int*   local  = (int*)(w + local_off);
  int*   offs   = (int*)(w + offs_off);
  int*   cursor = (int*)(w + cursor_off);
  int*   bsums  = (int*)(w + bsums_off);
  int*   permc  = (int*)(w + permc_off);
  float* permv  = (float*)(w + permv_off);

  // 1) histogram (counts re-zeroed every call: launch is self-contained)
  hipMemsetAsync(counts, 0, (size_t)N * 4, stream);
  const int grid_E = (E + block - 1) / block;
  hist_kernel<<<grid_E, block, 0, stream>>>(rows, counts, E);

  // 2) exclusive scan -> CSR row offsets (+ cursor working copy)
  scan_block_kernel<<<nb_scan, SCAN_BLOCK, 0, stream>>>(counts, local, bsums, N);
  scan_sums_kernel<<<1, 1024, 0, stream>>>(bsums, nb_scan);
  const int grid_N1 = (N + block - 1) / block;
  finalize_offsets_kernel<<<grid_N1, block, 0, stream>>>(local, bsums, offs, cursor, N, E);

  // 3) scatter edges into row-sorted (col,val) streams
  scatter_kernel<<<grid_E, block, 0, stream>>>(rows, cols, vals, cursor, permc, permv, E);

  // 4) gather SpMM with async-LDS staging; every out element written once
  const int grid_spmm = (N + ROWS_PER_BLOCK - 1) / ROWS_PER_BLOCK;
  spmm_csr_lds_kernel<<<grid_spmm, block, 0, stream>>>(offs, permc, permv, b, out, N);
}